// ALSHConv2d_87162066305060
// MI455X (gfx1250) — compile-verified
//
#include <hip/hip_runtime.h>
#include <hip/hip_bf16.h>
#include <math.h>

// ---- static problem config (matches reference) ----
#define OC      128
#define IC      64
#define HH      56
#define WW      56
#define BATCH   16
#define KK      3
#define SPAN    576           // KK*KK*IC
#define M_ALSH  5
#define TSIZE   8
#define RW      2.5f
#define HWPIX   (HH * WW)     // 3136
#define NPIX    (BATCH * HWPIX) // 50176
#define LDSPITCH 580          // 576 + 4 pad (576 % 64 == 0 -> conflicts otherwise)

typedef __attribute__((ext_vector_type(2))) float v2f;
typedef __attribute__((ext_vector_type(8))) float v8f;

// bucket = abs(fmod(int32(floor((v+b)/R)), T))  -- exact reference semantics:
// int cast truncates toward zero, C '%' matches fmod for int32, then abs.
__device__ __forceinline__ int bucket_of(float v, float bb) {
    float h = floorf((v + bb) / RW);
    int hi = (int)h;
    int m = hi % TSIZE;
    return m < 0 ? -m : m;
}

__global__ void alsh_init_kernel(int* __restrict__ counts) {
    int t = threadIdx.x;
    if (t < TSIZE) counts[t] = 0;
}

// One thread per output pixel: 576-tap LSH projection of the im2col column,
// Q-augmentation constant 0.5*sum(a[SPAN:]), histogram vote via global atomic.
__global__ void __launch_bounds__(256)
alsh_vote_kernel(const float* __restrict__ xin, const float* __restrict__ a,
                 const float* __restrict__ bptr, int* __restrict__ counts) {
    int n = blockIdx.x * blockDim.x + threadIdx.x;
    if (n >= NPIX) return;
    int bi  = n / HWPIX;
    int rem = n - bi * HWPIX;
    int y   = rem / WW;
    int x   = rem - y * WW;

    float v = 0.5f * (a[SPAN] + a[SPAN + 1] + a[SPAN + 2] + a[SPAN + 3] + a[SPAN + 4]);
    for (int c = 0; c < IC; ++c) {
        const float* xp = xin + (size_t)(bi * IC + c) * HWPIX;
        const float* ap = a + c * 9;
        #pragma unroll
        for (int kh = 0; kh < 3; ++kh) {
            int iy = y + kh - 1;
            #pragma unroll
            for (int kw = 0; kw < 3; ++kw) {
                int ix = x + kw - 1;
                float xv = ((unsigned)iy < (unsigned)HH && (unsigned)ix < (unsigned)WW)
                               ? xp[iy * WW + ix] : 0.0f;
                v = fmaf(ap[kh * 3 + kw], xv, v);
            }
        }
    }
    atomicAdd(&counts[bucket_of(v, bptr[0])], 1);
}

// One block, 128 threads (one per output channel): kernel-row hash with
// P-augmentation (norm powers ||k||^{2,4,8,16,32}), argmax bucket (first max,
// matching jnp.argmax), then factor = cnt>0 ? mask*OC/cnt : 1.
__global__ void __launch_bounds__(128)
alsh_factor_kernel(const float* __restrict__ w, const float* __restrict__ a,
                   const float* __restrict__ bptr, const int* __restrict__ counts,
                   float* __restrict__ factor) {
    __shared__ int s_bucket;
    __shared__ int s_cnt;
    int oc = threadIdx.x;
    if (oc == 0) {
        int best = 0, bestc = counts[0];
        for (int t = 1; t < TSIZE; ++t)
            if (counts[t] > bestc) { bestc = counts[t]; best = t; }
        s_bucket = best;
        s_cnt = 0;
    }
    __syncthreads();

    float n2 = 0.0f, dot = 0.0f;
    const float* wrow = w + (size_t)oc * SPAN;
    for (int s = 0; s < SPAN; ++s) {
        float wv = wrow[s];
        n2  = fmaf(wv, wv, n2);
        dot = fmaf(wv, a[s], dot);
    }
    float hk = dot;
    float p = n2;                       // powers: n2^1, n2^2, n2^4, n2^8, n2^16
    #pragma unroll
    for (int i = 0; i < M_ALSH; ++i) {
        hk = fmaf(p, a[SPAN + i], hk);
        p = p * p;
    }
    int msk = (bucket_of(hk, bptr[0]) == s_bucket) ? 1 : 0;
    if (msk) atomicAdd(&s_cnt, 1);
    __syncthreads();
    int cnt = s_cnt;
    factor[oc] = (cnt > 0) ? (msk ? ((float)OC / (float)cnt) : 0.0f) : 1.0f;
}

// WMMA GEMM: out[oc, n] = sum_s (w[oc,s]*factor[oc]) * P[s, n]
// Block = 256 threads = 8 wave32s; block covers OC=128 x 16 pixels.
// Each wave: one 16x16 tile, K-loop of 144 v_wmma_f32_16x16x4_f32.
__global__ void __launch_bounds__(256)
alsh_gemm_kernel(const float* __restrict__ xin, const float* __restrict__ w,
                 const float* __restrict__ factor, float* __restrict__ out) {
    __shared__ float ptile[16 * LDSPITCH];   // [n_local][s], padded pitch

    int n0 = blockIdx.x * 16;

    // Stage the 576 x 16 im2col tile into LDS (gathered fp32 loads; x sits in L2).
    for (int i = threadIdx.x; i < 16 * SPAN; i += 256) {
        int n_local = i / SPAN;
        int s = i - n_local * SPAN;
        int n = n0 + n_local;
        int bi  = n / HWPIX;
        int rem = n - bi * HWPIX;
        int y = rem / WW;
        int x = rem - y * WW;
        int c = s / 9;
        int r = s - c * 9;
        int kh = r / 3;
        int kw = r - kh * 3;
        int iy = y + kh - 1;
        int ix = x + kw - 1;
        float val = 0.0f;
        if ((unsigned)iy < (unsigned)HH && (unsigned)ix < (unsigned)WW)
            val = xin[((size_t)(bi * IC + c) * HH + iy) * WW + ix];
        ptile[n_local * LDSPITCH + s] = val;
    }
    __syncthreads();

    int wave = threadIdx.x >> 5;     // 0..7 -> M tile
    int lane = threadIdx.x & 31;
    int half = lane >> 4;            // 0: K pair {0,1}; 1: K pair {2,3}
    int l15  = lane & 15;
    int m0   = wave * 16;

    // A fragment source: row m0+l15 of W, scaled by factor (ISA A-16x4 layout).
    int arow = m0 + l15;
    float f = factor[arow];
    const float* wrow = w + (size_t)arow * SPAN + half * 2;

    v8f acc = {};
    for (int k = 0; k < SPAN; k += 4) {
        v2f aa;
        aa.x = wrow[k]     * f;      // A[M=arow][K=k+2*half]
        aa.y = wrow[k + 1] * f;      // A[M=arow][K=k+2*half+1]
        const float* bp = &ptile[l15 * LDSPITCH + k + half * 2];
        v2f bb;
        bb.x = bp[0];                // B[K=k+2*half  ][N=l15]
        bb.y = bp[1];                // B[K=k+2*half+1][N=l15]
        acc = __builtin_amdgcn_wmma_f32_16x16x4_f32(
            /*neg_a=*/false, aa, /*neg_b=*/false, bb,
            /*c_mod=*/(short)0, acc, /*reuse_a=*/false, /*reuse_b=*/false);
    }

    // D layout: VGPR r, lanes 0-15 -> M=r, N=lane; lanes 16-31 -> M=r+8, N=lane-16.
    int oc_base = m0 + half * 8;
    int n = n0 + l15;
    int bi  = n / HWPIX;
    int pos = n - bi * HWPIX;
    float* obase = out + (size_t)bi * OC * HWPIX + pos;
    #pragma unroll
    for (int r = 0; r < 8; ++r)
        obase[(size_t)(oc_base + r) * HWPIX] = acc[r];
}

extern "C" void kernel_launch(void* const* d_in, const int* in_sizes, int n_in,
                              void* d_out, int out_size, void* d_ws, size_t ws_size,
                              hipStream_t stream) {
    const float* x    = (const float*)d_in[0];   // [16,64,56,56]
    const float* kern = (const float*)d_in[1];   // [128,576]
    const float* a    = (const float*)d_in[2];   // [581]
    const float* b    = (const float*)d_in[3];   // scalar
    (void)in_sizes; (void)n_in; (void)out_size; (void)ws_size;

    int*   counts = (int*)d_ws;                       // 8 ints
    float* factor = (float*)((char*)d_ws + 64);       // 128 floats
    float* out    = (float*)d_out;                    // [16,128,56,56]

    hipLaunchKernelGGL(alsh_init_kernel, dim3(1), dim3(32), 0, stream, counts);
    hipLaunchKernelGGL(alsh_vote_kernel, dim3((NPIX + 255) / 256), dim3(256), 0, stream,
                       x, a, b, counts);
    hipLaunchKernelGGL(alsh_factor_kernel, dim3(1), dim3(128), 0, stream,
                       kern, a, b, counts, factor);
    hipLaunchKernelGGL(alsh_gemm_kernel, dim3(NPIX / 16), dim3(256), 0, stream,
                       x, kern, factor, out);
}